// GAT_attention_single_12910671692373
// MI455X (gfx1250) — compile-verified
//
#include <hip/hip_runtime.h>
#include <math.h>

typedef __attribute__((ext_vector_type(2))) float v2f;
typedef __attribute__((ext_vector_type(8))) float v8f;

#define LN_EPS   1e-5f
#define NSLOPE   0.01f
#define TA_S     68     // 16x64 tile row stride (pad: 68%64=4 -> conflict-free column reads)
#define P_S      260    // 16x256 row stride   (pad: 260%64=4 -> conflict-free column reads)

__device__ __forceinline__ float wave_sum(float v) {
#pragma unroll
  for (int o = 16; o > 0; o >>= 1) v += __shfl_xor(v, o, 32);
  return v;
}
__device__ __forceinline__ float wave_max(float v) {
#pragma unroll
  for (int o = 16; o > 0; o >>= 1) v = fmaxf(v, __shfl_xor(v, o, 32));
  return v;
}
__device__ __forceinline__ float leaky(float x) { return x >= 0.0f ? x : NSLOPE * x; }

// D = A(16x4,f32) * B(4x16,f32) + C ; fp32 WMMA keeps reference precision.
__device__ __forceinline__ v8f wmma4(v2f a, v2f b, v8f c) {
  return __builtin_amdgcn_wmma_f32_16x16x4_f32(false, a, false, b, (short)0, c, false, false);
}

// ---------------------------------------------------------------------------
// Kernel 1: fold Wa into the attention vectors.
// vq = aq @ Wa, vk = ak @ Wa, vi = ai @ Wa ; cq=ba.aq, ck=ba.ak, ci=ba.ai
// vbuf: [0..63]=vq [64..127]=vk [128..191]=vi [192]=cq [193]=ck [194]=ci
// ---------------------------------------------------------------------------
__global__ void gat_prep_vecs(const float* __restrict__ Wa,
                              const float* __restrict__ ba,
                              const float* __restrict__ a_w,
                              float* __restrict__ vbuf) {
  const int k = threadIdx.x;  // 64 threads
  if (k < 64) {
    float q = 0.f, kk = 0.f, ii = 0.f;
    for (int m = 0; m < 64; ++m) {
      const float w = Wa[m * 64 + k];
      q  += a_w[m]        * w;
      kk += a_w[64 + m]   * w;
      ii += a_w[128 + m]  * w;
    }
    vbuf[k]        = q;
    vbuf[64 + k]   = kk;
    vbuf[128 + k]  = ii;
  }
  if (threadIdx.x == 0) {
    float cq = 0.f, ck = 0.f, ci = 0.f;
    for (int m = 0; m < 64; ++m) {
      const float b = ba[m];
      cq += b * a_w[m];
      ck += b * a_w[64 + m];
      ci += b * a_w[128 + m];
    }
    vbuf[192] = cq; vbuf[193] = ck; vbuf[194] = ci;
  }
}

// ---------------------------------------------------------------------------
// Kernel 2: per batch — LayerNorm rows, ua = uid0*iatt, ua2 = ua*ua,
// scores s_q, s_k, s_i, and output row 0 (leaky(uid0*iid0)).
// One block per batch, 256 threads = 8 waves. Wave handles one row at a time,
// 2 elements per lane (D=64, wave32).
// ---------------------------------------------------------------------------
__global__ void gat_prep_batch(const float* __restrict__ emb,
                               const float* __restrict__ ln_g,
                               const float* __restrict__ ln_b,
                               const float* __restrict__ vbuf,
                               float* __restrict__ ua,
                               float* __restrict__ ua2,
                               float* __restrict__ sq,
                               float* __restrict__ sk,
                               float* __restrict__ si,
                               float* __restrict__ out) {
  const int b    = blockIdx.x;
  const int lane = threadIdx.x & 31;
  const int wave = threadIdx.x >> 5;
  __shared__ float uid_ln[64];
  __shared__ float iid_ln[64];
  const float* E = emb + (size_t)b * 258 * 64;

  const float g0 = ln_g[lane], g1 = ln_g[lane + 32];
  const float b0 = ln_b[lane], b1 = ln_b[lane + 32];

  // LayerNorm of one 64-wide row, 2 elems/lane.
  auto ln_row = [&](const float* row, float& o0, float& o1) {
    const float x0 = row[lane], x1 = row[lane + 32];
    const float mu = wave_sum(x0 + x1) * (1.0f / 64.0f);
    const float d0 = x0 - mu, d1 = x1 - mu;
    const float var = wave_sum(d0 * d0 + d1 * d1) * (1.0f / 64.0f);
    const float inv = rsqrtf(var + LN_EPS);
    o0 = d0 * inv * g0 + b0;
    o1 = d1 * inv * g1 + b1;
  };

  if (wave == 0) { float o0, o1; ln_row(E,       o0, o1); uid_ln[lane] = o0; uid_ln[lane + 32] = o1; }
  if (wave == 1) { float o0, o1; ln_row(E + 64,  o0, o1); iid_ln[lane] = o0; iid_ln[lane + 32] = o1; }
  __syncthreads();

  if (wave == 0) {
    // output row 0: leaky(uid0 * iid0)
    const float u0 = uid_ln[lane] * iid_ln[lane];
    const float u1 = uid_ln[lane + 32] * iid_ln[lane + 32];
    float* o = out + (size_t)b * 257 * 64;
    o[lane]      = leaky(u0);
    o[lane + 32] = leaky(u1);
    // s_i = iid_ln . vi + ci
    const float s = wave_sum(iid_ln[lane] * vbuf[128 + lane] +
                             iid_ln[lane + 32] * vbuf[128 + lane + 32]);
    if (lane == 0) si[b] = s + vbuf[194];
  }

  const float vq0 = vbuf[lane],      vq1 = vbuf[lane + 32];
  const float vk0 = vbuf[64 + lane], vk1 = vbuf[64 + lane + 32];
  const float uu0 = uid_ln[lane],    uu1 = uid_ln[lane + 32];

  for (int r = wave; r < 256; r += 8) {
    float o0, o1;
    ln_row(E + (size_t)(2 + r) * 64, o0, o1);
    const float u0 = o0 * uu0, u1 = o1 * uu1;
    float* up  = ua  + ((size_t)b * 256 + r) * 64;
    float* u2p = ua2 + ((size_t)b * 256 + r) * 64;
    up[lane] = u0;            up[lane + 32] = u1;
    u2p[lane] = u0 * u0;      u2p[lane + 32] = u1 * u1;
    const float dq = wave_sum(u0 * vq0 + u1 * vq1);
    const float dk = wave_sum(u0 * vk0 + u1 * vk1);
    if (lane == 0) {
      sq[b * 256 + r] = dq + vbuf[192];
      sk[b * 256 + r] = dk + vbuf[193];
    }
  }
}

// ---------------------------------------------------------------------------
// Kernel 3: one block per (batch, 16-row tile). 256 threads = 8 waves.
// Stage A: WMMA Gram tiles M = A·uaT, S = A2·ua2T (each wave: 2 j-tiles,
//          16 k-steps of v_wmma_f32_16x16x4_f32 per accumulator).
// Stage B: row softmax of leaky(logits), build W_ij = alpha*inv, c_i.
// Stage C: WMMA att_core = W(16x256)·ua(256x64), fused LN-affine epilogue.
// ---------------------------------------------------------------------------
__global__ void gat_attn(const float* __restrict__ ua,
                         const float* __restrict__ ua2,
                         const float* __restrict__ sq,
                         const float* __restrict__ sk,
                         const float* __restrict__ si,
                         const float* __restrict__ a_b,
                         const float* __restrict__ ln_g,
                         const float* __restrict__ ln_b,
                         float* __restrict__ out) {
  const int b  = blockIdx.x >> 4;
  const int t  = blockIdx.x & 15;
  const int i0 = t * 16;

  const int lane  = threadIdx.x & 31;
  const int wave  = threadIdx.x >> 5;
  const int mrow  = lane & 15;   // fast index (A row / B col)
  const int khalf = lane >> 4;   // which K-half this lane holds

  __shared__ float sA [16 * TA_S];   // ua  rows i0..i0+15
  __shared__ float sA2[16 * TA_S];   // ua^2 rows
  __shared__ float sP [16 * P_S];    // logits -> then W_ij = alpha*inv
  __shared__ float sMI[16 * P_S];    // mean*inv
  __shared__ float sIV[16 * P_S];    // inv
  __shared__ float sKs[256];
  __shared__ float sQs[16];
  __shared__ float sC [16];

  const float* uab  = ua  + (size_t)b * 256 * 64;
  const float* ua2b = ua2 + (size_t)b * 256 * 64;

  { // cooperative tile + score loads
    const int row = threadIdx.x >> 4;        // 0..15
    const int col = (threadIdx.x & 15) * 4;  // 0..60
    const float4 va = *(const float4*)(uab  + (size_t)(i0 + row) * 64 + col);
    const float4 vb = *(const float4*)(ua2b + (size_t)(i0 + row) * 64 + col);
    *(float4*)&sA [row * TA_S + col] = va;
    *(float4*)&sA2[row * TA_S + col] = vb;
    sKs[threadIdx.x] = sk[b * 256 + threadIdx.x];
    if (threadIdx.x < 16) sQs[threadIdx.x] = sq[b * 256 + i0 + threadIdx.x];
  }
  __syncthreads();

  const float s_i = si[b];
  const float ab  = a_b[0];

  // ---- Stage A: Gram matrices via fp32 WMMA ----
  for (int jj = 0; jj < 2; ++jj) {
    const int j0 = (wave + jj * 8) * 16;
    v8f accM = {0.f, 0.f, 0.f, 0.f, 0.f, 0.f, 0.f, 0.f};
    v8f accS = {0.f, 0.f, 0.f, 0.f, 0.f, 0.f, 0.f, 0.f};
    const float* browM = uab  + (size_t)(j0 + mrow) * 64;
    const float* browS = ua2b + (size_t)(j0 + mrow) * 64;
#pragma unroll
    for (int s = 0; s < 16; ++s) {
      const int k = s * 4 + 2 * khalf;
      const v2f aM = *(const v2f*)&sA[mrow * TA_S + k];
      const v2f bM = *(const v2f*)(browM + k);
      accM = wmma4(aM, bM, accM);
      const v2f aS = *(const v2f*)&sA2[mrow * TA_S + k];
      const v2f bS = *(const v2f*)(browS + k);
      accS = wmma4(aS, bS, accS);
    }
#pragma unroll
    for (int v = 0; v < 8; ++v) {
      const int il = v + 8 * khalf;   // local i
      const int j  = j0 + mrow;       // global j
      const float mean = accM[v] * (1.0f / 64.0f);
      const float e2   = accS[v] * (1.0f / 64.0f);
      const float var  = e2 - mean * mean;
      const float inv  = rsqrtf(var + LN_EPS);
      sP [il * P_S + j] = leaky(sQs[il] + sKs[j] + s_i + ab);
      sMI[il * P_S + j] = mean * inv;
      sIV[il * P_S + j] = inv;
    }
  }
  __syncthreads();

  // ---- Stage B: softmax over j, build W and c ----
  for (int rr = 0; rr < 2; ++rr) {
    const int i = wave + rr * 8;
    float vals[8];
    float mx = -3.402823466e38f;
#pragma unroll
    for (int q = 0; q < 8; ++q) {
      vals[q] = sP[i * P_S + lane + 32 * q];
      mx = fmaxf(mx, vals[q]);
    }
    mx = wave_max(mx);
    float ssum = 0.f;
#pragma unroll
    for (int q = 0; q < 8; ++q) { vals[q] = __expf(vals[q] - mx); ssum += vals[q]; }
    ssum = wave_sum(ssum);
    const float isum = 1.0f / ssum;
    float cpart = 0.f;
#pragma unroll
    for (int q = 0; q < 8; ++q) {
      const int j = lane + 32 * q;
      const float alpha = vals[q] * isum;
      cpart += alpha * sMI[i * P_S + j];
      sP[i * P_S + j] = alpha * sIV[i * P_S + j];   // W_ij
    }
    cpart = wave_sum(cpart);
    if (lane == 0) sC[i] = -cpart;
  }
  __syncthreads();

  // ---- Stage C: att_core = W(16x256) @ ua(256x64), waves 0..3 own d-tiles ----
  if (wave < 4) {
    const int n0 = wave * 16;
    v8f acc = {0.f, 0.f, 0.f, 0.f, 0.f, 0.f, 0.f, 0.f};
    for (int s = 0; s < 64; ++s) {
      const int k = s * 4 + 2 * khalf;                  // j index of this lane's regs
      const v2f av = *(const v2f*)&sP[mrow * P_S + k];  // A[m][k],A[m][k+1]
      v2f bv;
      bv.x = uab[(size_t)(k)     * 64 + n0 + mrow];     // B[k][n]
      bv.y = uab[(size_t)(k + 1) * 64 + n0 + mrow];     // B[k+1][n]
      acc = wmma4(av, bv, acc);
    }
    const int d  = n0 + mrow;
    const float g  = ln_g[d];
    const float bb = ln_b[d];
#pragma unroll
    for (int v = 0; v < 8; ++v) {
      const int il = v + 8 * khalf;
      const float val = sA[il * TA_S + d] * g * acc[v] + g * sC[il] + bb;
      out[((size_t)b * 257 + 1 + i0 + il) * 64 + d] = leaky(val);
    }
  }
}

// ---------------------------------------------------------------------------
extern "C" void kernel_launch(void* const* d_in, const int* in_sizes, int n_in,
                              void* d_out, int out_size, void* d_ws, size_t ws_size,
                              hipStream_t stream) {
  const float* emb  = (const float*)d_in[0];  // (16,258,64)
  const float* Wa   = (const float*)d_in[1];  // (64,64)
  const float* ba   = (const float*)d_in[2];  // (64,)
  const float* a_w  = (const float*)d_in[3];  // (192,)
  const float* a_b  = (const float*)d_in[4];  // (1,)
  const float* ln_g = (const float*)d_in[5];  // (64,)
  const float* ln_b = (const float*)d_in[6];  // (64,)
  float* out = (float*)d_out;                 // (16,257,64)

  float* ws   = (float*)d_ws;
  float* ua   = ws;                        // 16*256*64
  float* ua2  = ua  + 16 * 256 * 64;       // 16*256*64
  float* sq   = ua2 + 16 * 256 * 64;       // 16*256
  float* sk   = sq  + 16 * 256;            // 16*256
  float* si   = sk  + 16 * 256;            // 16
  float* vbuf = si  + 16;                  // 195

  gat_prep_vecs <<<1,   64,  0, stream>>>(Wa, ba, a_w, vbuf);
  gat_prep_batch<<<16,  256, 0, stream>>>(emb, ln_g, ln_b, vbuf, ua, ua2, sq, sk, si, out);
  gat_attn      <<<256, 256, 0, stream>>>(ua, ua2, sq, sk, si, a_b, ln_g, ln_b, out);
}